// Model_70884140253869
// MI455X (gfx1250) — compile-verified
//
#include <hip/hip_runtime.h>
#include <math.h>

// ---------------------------------------------------------------------------
// MI455X (gfx1250) implementation of the dynamic-graph rec model.
// fp32 throughout (reference is fp32). GEMMs use V_WMMA_F32_16X16X4_F32,
// one wave per 16x64 output strip (4 accumulator tiles, A-fragment reuse).
// GNN scatter uses global f32 atomics (targets fit in 192MB L2).
// All reductions are wave32.
// ---------------------------------------------------------------------------

typedef __attribute__((ext_vector_type(2))) float v2f;
typedef __attribute__((ext_vector_type(8))) float v8f;

__device__ __forceinline__ float lky(float x)  { return fmaxf(x, 0.01f * x); }   // source leakyRelu
__device__ __forceinline__ float sigm(float x) { return 1.0f / (1.0f + __expf(-x)); }

__device__ __forceinline__ float wave_sum(float v) {
#pragma unroll
    for (int off = 16; off > 0; off >>= 1) v += __shfl_xor(v, off, 32);
    return v;
}

__device__ __forceinline__ void atomic_add_f(float* p, float v) {
    __hip_atomic_fetch_add(p, v, __ATOMIC_RELAXED, __HIP_MEMORY_SCOPE_AGENT);
}

// LayerNorm of a 64-wide row held as (v0 = x[lane], v1 = x[lane+32]) in a wave.
__device__ __forceinline__ void ln_pair(float v0, float v1,
                                        const float* __restrict__ g,
                                        const float* __restrict__ b,
                                        int lane, float& o0, float& o1) {
    float mean = wave_sum(v0 + v1) * (1.0f / 64.0f);
    float d0 = v0 - mean, d1 = v1 - mean;
    float var = wave_sum(d0 * d0 + d1 * d1) * (1.0f / 64.0f);
    float r = rsqrtf(var + 1e-3f);                 // keras eps = 1e-3
    o0 = d0 * r * g[lane]      + b[lane];
    o1 = d1 * r * g[lane + 32] + b[lane + 32];
}

// ------------------------------ utility kernels ----------------------------

__global__ void k_zero(float* __restrict__ p, long n) {
    long i = (long)blockIdx.x * blockDim.x + threadIdx.x;
    if (i < n) p[i] = 0.0f;
}

__global__ void k_copy2(float* __restrict__ d1, float* __restrict__ d2,
                        const float* __restrict__ s, long n) {
    long i = (long)blockIdx.x * blockDim.x + threadIdx.x;
    if (i < n) { float v = s[i]; d1[i] = v; d2[i] = v; }
}

// ------------------------------ GNN ----------------------------------------
// acc[tgt[e]] += emb[src[e]]  (64 floats per edge, 16 threads/edge, float4)
__global__ void k_scatter(float* __restrict__ acc, const float* __restrict__ emb,
                          const int* __restrict__ src, const int* __restrict__ tgt, long E) {
    long t = (long)blockIdx.x * blockDim.x + threadIdx.x;
    long e = t >> 4;
    if (e >= E) return;
    int q = (int)(t & 15) * 4;
    const float4 v = *(const float4*)(emb + (long)src[e] * 64 + q);
    float* d = acc + (long)tgt[e] * 64 + q;
    atomic_add_f(d + 0, v.x);
    atomic_add_f(d + 1, v.y);
    atomic_add_f(d + 2, v.z);
    atomic_add_f(d + 3, v.w);
}

// cur = leaky(acc) + cur ; sum += cur
__global__ void k_gnn_update(float* __restrict__ cur, float* __restrict__ sum,
                             const float* __restrict__ acc, long n) {
    long i = (long)blockIdx.x * blockDim.x + threadIdx.x;
    if (i < n) { float v = lky(acc[i]) + cur[i]; cur[i] = v; sum[i] += v; }
}

// ------------------------------ WMMA GEMM ----------------------------------
// K-panel of V_WMMA_F32_16X16X4_F32: accumulates A[16,K] @ B[K,64] into four
// 16x16 tiles. ap pre-offset to (row * lda + 2*hf); bp to (2*hf)*ldb + col.
// A 16x4 VGPR layout: lanes 0-15 hold K={k,k+1}, lanes 16-31 hold K={k+2,k+3};
// B mirrors it with lanes giving N. C tile: row = v + 8*(lane/16), col = lane&15.
__device__ __forceinline__ void gemm_panel(const float* __restrict__ ap,
                                           const float* __restrict__ bp,
                                           int ldb, int K, v8f acc[4]) {
    for (int k0 = 0; k0 < K; k0 += 4) {
        v2f a;
        a[0] = ap[0];
        a[1] = ap[1];
#pragma unroll
        for (int t = 0; t < 4; ++t) {
            v2f b;
            b[0] = bp[t * 16];
            b[1] = bp[t * 16 + ldb];
            acc[t] = __builtin_amdgcn_wmma_f32_16x16x4_f32(
                /*neg_a=*/false, a, /*neg_b=*/false, b,
                /*c_mod=*/(short)0, acc[t], /*reuse_a=*/false, /*reuse_b=*/false);
        }
        ap += 4;
        bp += 4 * (size_t)ldb;
    }
}

// C[M,N] = A1[M,K] @ B1[K,N] (+ A2[M,K] @ B2[K,N]) (+ bias[N]).
// One wave per 16x64 output strip. grid = (N/64, M/16).
// M mult of 16, N mult of 64, K mult of 4. Row-major. A2 may be null.
__global__ void __launch_bounds__(32)
k_gemm_wmma(float* __restrict__ C,
            const float* __restrict__ A1, const float* __restrict__ B1,
            const float* __restrict__ A2, const float* __restrict__ B2,
            const float* __restrict__ bias,
            int K, int lda1, int ldb1, int lda2, int ldb2, int ldc) {
    const int lane = threadIdx.x & 31;
    const int hf   = lane >> 4;         // which 16-lane half
    const int l16  = lane & 15;
    const int row0 = blockIdx.y * 16;
    const int col0 = blockIdx.x * 64;

    v8f acc[4];
#pragma unroll
    for (int t = 0; t < 4; ++t) {
        float bv = bias ? bias[col0 + t * 16 + l16] : 0.0f;
#pragma unroll
        for (int v = 0; v < 8; ++v) acc[t][v] = bv;
    }

    gemm_panel(A1 + (size_t)(row0 + l16) * lda1 + 2 * hf,
               B1 + (size_t)(2 * hf) * ldb1 + col0 + l16, ldb1, K, acc);
    if (A2)
        gemm_panel(A2 + (size_t)(row0 + l16) * lda2 + 2 * hf,
                   B2 + (size_t)(2 * hf) * ldb2 + col0 + l16, ldb2, K, acc);

    float* cp = C + (size_t)(row0 + 8 * hf) * ldc + col0 + l16;
#pragma unroll
    for (int t = 0; t < 4; ++t)
#pragma unroll
        for (int v = 0; v < 8; ++v)
            cp[(size_t)v * ldc + t * 16] = acc[t][v];
}

// ------------------------------ LSTM gates ---------------------------------
// z[N,256] -> (h,c) update; h also written to seq_out[:, t, :]  ([N,4,64])
__global__ void k_lstm_gates(const float* __restrict__ z, float* __restrict__ h,
                             float* __restrict__ c, float* __restrict__ seq_out,
                             int N, int t) {
    long idx = (long)blockIdx.x * blockDim.x + threadIdx.x;
    if (idx >= (long)N * 64) return;
    long r = idx >> 6;
    int d = (int)(idx & 63);
    const float* zr = z + r * 256;
    float iv = sigm(zr[d]);
    float fv = sigm(zr[64 + d]);
    float gv = tanhf(zr[128 + d]);
    float ov = sigm(zr[192 + d]);
    float cv = fv * c[idx] + iv * gv;
    float hv = ov * tanhf(cv);
    c[idx] = cv;
    h[idx] = hv;
    seq_out[r * 256 + t * 64 + d] = hv;
}

// ---------------------- fused LN + 4-step MHA + mean -----------------------
// X:[N,4,64] -> outF:[N,64]. One wave per row; 4 rows / 128-thread block.
__global__ void __launch_bounds__(128)
k_ln_mha_mean(const float* __restrict__ X,
              const float* __restrict__ lg, const float* __restrict__ lb,
              const float* __restrict__ Wq, const float* __restrict__ Wk,
              const float* __restrict__ Wv, float* __restrict__ outF, int N) {
    __shared__ float s_xln[4][4][64];
    __shared__ float s_q[4][4][64];
    __shared__ float s_k[4][4][64];
    __shared__ float s_v[4][4][64];
    __shared__ float s_o[4][4][64];

    int w = threadIdx.x >> 5, lane = threadIdx.x & 31;
    long n = (long)blockIdx.x * 4 + w;
    bool active = (n < N);

    // LayerNorm each of the 4 snapshot vectors
    for (int s = 0; s < 4; ++s) {
        float v0 = 0.f, v1 = 0.f;
        if (active) {
            v0 = X[(n * 4 + s) * 64 + lane];
            v1 = X[(n * 4 + s) * 64 + lane + 32];
        }
        float o0, o1;
        ln_pair(v0, v1, lg, lb, lane, o0, o1);
        if (active) { s_xln[w][s][lane] = o0; s_xln[w][s][lane + 32] = o1; }
    }
    __syncthreads();

    // Q,K,V projections: each lane computes 2 columns for all 4 steps
    if (active) {
        int c0 = lane, c1 = lane + 32;
        for (int s = 0; s < 4; ++s) {
            float aq0 = 0, aq1 = 0, ak0 = 0, ak1 = 0, av0 = 0, av1 = 0;
            for (int k = 0; k < 64; ++k) {
                float x = s_xln[w][s][k];
                aq0 += x * Wq[k * 64 + c0]; aq1 += x * Wq[k * 64 + c1];
                ak0 += x * Wk[k * 64 + c0]; ak1 += x * Wk[k * 64 + c1];
                av0 += x * Wv[k * 64 + c0]; av1 += x * Wv[k * 64 + c1];
            }
            s_q[w][s][c0] = aq0; s_q[w][s][c1] = aq1;
            s_k[w][s][c0] = ak0; s_k[w][s][c1] = ak1;
            s_v[w][s][c0] = av0; s_v[w][s][c1] = av1;
        }
    }
    __syncthreads();

    // attention: 16 (head, query) pairs on lanes 0..15 (H=4, S=4, DH=16)
    if (active && lane < 16) {
        int hh = lane >> 2, s = lane & 3;
        const float* qp = &s_q[w][s][hh * 16];
        float sc[4], mx = -1e30f;
        for (int s2 = 0; s2 < 4; ++s2) {
            float d = 0;
            for (int j = 0; j < 16; ++j) d += qp[j] * s_k[w][s2][hh * 16 + j];
            sc[s2] = d * 0.25f;                      // 1/sqrt(DH), DH=16
            mx = fmaxf(mx, sc[s2]);
        }
        float psum = 0;
        for (int s2 = 0; s2 < 4; ++s2) { sc[s2] = __expf(sc[s2] - mx); psum += sc[s2]; }
        float inv = 1.0f / psum;
        for (int j = 0; j < 16; ++j) {
            float o = 0;
            for (int s2 = 0; s2 < 4; ++s2) o += sc[s2] * s_v[w][s2][hh * 16 + j];
            s_o[w][s][hh * 16 + j] = o * inv;
        }
    }
    __syncthreads();

    // mean over the 4 steps
    if (active) {
        float m0 = 0, m1 = 0;
        for (int s = 0; s < 4; ++s) { m0 += s_o[w][s][lane]; m1 += s_o[w][s][lane + 32]; }
        outF[n * 64 + lane]      = m0 * 0.25f;
        outF[n * 64 + lane + 32] = m1 * 0.25f;
    }
}

// ------------- masked sequence pooling + LN(s1)+LN(s2) ---------------------
__global__ void __launch_bounds__(128)
k_seq_pool(const float* __restrict__ mask, const int* __restrict__ seq,
           const float* __restrict__ FI, const float* __restrict__ pos,
           const float* __restrict__ g2, const float* __restrict__ b2,
           const float* __restrict__ g3, const float* __restrict__ b3,
           float* __restrict__ att, int Bn) {
    int w = threadIdx.x >> 5, lane = threadIdx.x & 31;
    int b = blockIdx.x * 4 + w;
    if (b >= Bn) return;
    float s1_0 = 0, s1_1 = 0, s2_0 = 0, s2_1 = 0;
    for (int l = 0; l < 100; ++l) {
        float m = mask[b * 100 + l];
        long it = seq[b * 100 + l];
        s1_0 += m * FI[it * 64 + lane];
        s1_1 += m * FI[it * 64 + lane + 32];
        s2_0 += m * pos[l * 64 + lane];
        s2_1 += m * pos[l * 64 + lane + 32];
    }
    float a0, a1, c0, c1;
    ln_pair(s1_0, s1_1, g2, b2, lane, a0, a1);
    ln_pair(s2_0, s2_1, g3, b3, lane, c0, c1);
    att[(long)b * 64 + lane]      = a0 + c0;
    att[(long)b * 64 + lane + 32] = a1 + c1;
}

__global__ void __launch_bounds__(128)
k_ln_rows(const float* __restrict__ x, const float* __restrict__ g,
          const float* __restrict__ b, float* __restrict__ y, int N) {
    int w = threadIdx.x >> 5, lane = threadIdx.x & 31;
    long r = (long)blockIdx.x * 4 + w;
    if (r >= N) return;
    float o0, o1;
    ln_pair(x[r * 64 + lane], x[r * 64 + lane + 32], g, b, lane, o0, o1);
    y[r * 64 + lane] = o0;
    y[r * 64 + lane + 32] = o1;
}

__global__ void k_leaky_add(float* __restrict__ y, const float* __restrict__ t, long n) {
    long i = (long)blockIdx.x * blockDim.x + threadIdx.x;
    if (i < n) y[i] = lky(t[i]) + y[i];
}

// ------------------------------ predictions --------------------------------
__global__ void __launch_bounds__(128)
k_preds(const float* __restrict__ FU, const float* __restrict__ FI,
        const float* __restrict__ att, const int* __restrict__ uids,
        const int* __restrict__ iids, const int* __restrict__ ulocs,
        float* __restrict__ out, int P) {
    int w = threadIdx.x >> 5, lane = threadIdx.x & 31;
    int p = blockIdx.x * 4 + w;
    if (p >= P) return;
    long u = uids[p], it = iids[p], bu = ulocs[p];
    float a = 0;
#pragma unroll
    for (int rep = 0; rep < 2; ++rep) {
        int d = lane + rep * 32;
        float fi = FI[it * 64 + d];
        a += FU[u * 64 + d] * fi + lky(att[bu * 64 + d]) * fi;
    }
    a = wave_sum(a);
    if (lane == 0) out[p] = a;
}

// ------------------------------ SSL ----------------------------------------
// w[u] = sigmoid( leaky_relu_0.2( concat(fu*uv, fu, uv) @ W1 + b1 ) @ W2 + b2 )
// lane = hidden unit (SSLD == 32 == wave width)
__global__ void __launch_bounds__(128)
k_ssl_w(const float* __restrict__ FU, const float* __restrict__ UVk,
        const float* __restrict__ W1, const float* __restrict__ b1,
        const float* __restrict__ W2, const float* __restrict__ b2,
        float* __restrict__ wout, int Un) {
    int w = threadIdx.x >> 5, lane = threadIdx.x & 31;
    long u = (long)blockIdx.x * 4 + w;
    if (u >= Un) return;
    float acc = b1[lane];
    for (int k = 0; k < 64; ++k) {
        float fu = FU[u * 64 + k];
        float uv = UVk[u * 64 + k];
        acc += (fu * uv) * W1[k * 32 + lane]
             + fu * W1[(64 + k) * 32 + lane]
             + uv * W1[(128 + k) * 32 + lane];
    }
    float hid = acc > 0.0f ? acc : 0.2f * acc;      // jax leaky_relu slope 0.2
    float t = wave_sum(hid * W2[lane]);
    if (lane == 0) wout[u] = sigm(t + b2[0]);
}

__global__ void __launch_bounds__(128)
k_ssl_pair(const float* __restrict__ FU, const float* __restrict__ FI,
           const float* __restrict__ UVk, const float* __restrict__ IVk,
           const float* __restrict__ wbuf, const int* __restrict__ su,
           const int* __restrict__ si, float* __restrict__ loss, int half) {
    int w = threadIdx.x >> 5, lane = threadIdx.x & 31;
    int s = blockIdx.x * 4 + w;
    if (s >= half) return;
    long up = su[s], ip = si[s], un = su[s + half], in2 = si[s + half];
    float Sp = 0, Sn = 0, Pp = 0, Pn = 0;
#pragma unroll
    for (int rep = 0; rep < 2; ++rep) {
        int d = lane + rep * 32;
        Sp += lky(FU[up * 64 + d] * FI[ip * 64 + d]);
        Sn += lky(FU[un * 64 + d] * FI[in2 * 64 + d]);
        Pp += lky(UVk[up * 64 + d] * IVk[ip * 64 + d]);
        Pn += lky(UVk[un * 64 + d] * IVk[in2 * 64 + d]);
    }
    Sp = wave_sum(Sp); Sn = wave_sum(Sn); Pp = wave_sum(Pp); Pn = wave_sum(Pn);
    if (lane == 0) {
        float Sf = wbuf[up] * Sp - wbuf[un] * Sn;
        float l = 1.0f - Sf * (Pp - Pn);
        if (l > 0.0f) atomic_add_f(loss, l);
    }
}

// ---------------------------------------------------------------------------
// Host-side orchestration.
// Input flattening (setup_inputs insertion order, params pytree first):
//  0 user_emb  1 item_emb
//  2 lstm0.Wx  3 lstm0.Wh  4 lstm0.b   5 lstm1.Wx  6 lstm1.Wh  7 lstm1.b
//  8 att0.Wq   9 att0.Wk  10 att0.Wv  11 att1.Wq  12 att1.Wk  13 att1.Wv
// 14 as0.Wq   15 as0.Wk   16 as0.Wv   17 as1.Wq   18 as1.Wk   19 as1.Wv
// 20..29 ln[0..4].{g,b}   30 pos_emb  31 fc1.W 32 fc1.b 33 fc2.W 34 fc2.b
// 35 adj_src 36 adj_tgt 37 tp_src 38 tp_tgt 39 mask 40 sequence
// 41 uids 42 iids 43 uLocs_seq 44 suids 45 siids
// ---------------------------------------------------------------------------
extern "C" void kernel_launch(void* const* d_in, const int* in_sizes, int n_in,
                              void* d_out, int out_size, void* d_ws, size_t ws_size,
                              hipStream_t stream) {
    (void)n_in; (void)out_size; (void)ws_size;
    constexpr int G = 4, U = 100000, I = 40000, D = 64, B = 2048, P = 2048;
    constexpr int SAMP = 4096, HALF = 2048, ZN = 4 * D;

    const float* user_emb = (const float*)d_in[0];
    const float* item_emb = (const float*)d_in[1];
    const float* l0Wx = (const float*)d_in[2];
    const float* l0Wh = (const float*)d_in[3];
    const float* l0b  = (const float*)d_in[4];
    const float* l1Wx = (const float*)d_in[5];
    const float* l1Wh = (const float*)d_in[6];
    const float* l1b  = (const float*)d_in[7];
    const float* a0Wq = (const float*)d_in[8];
    const float* a0Wk = (const float*)d_in[9];
    const float* a0Wv = (const float*)d_in[10];
    const float* a1Wq = (const float*)d_in[11];
    const float* a1Wk = (const float*)d_in[12];
    const float* a1Wv = (const float*)d_in[13];
    const float* asWv[2] = { (const float*)d_in[16], (const float*)d_in[19] };
    const float* ln_g[5]; const float* ln_b[5];
    for (int i = 0; i < 5; ++i) {
        ln_g[i] = (const float*)d_in[20 + 2 * i];
        ln_b[i] = (const float*)d_in[21 + 2 * i];
    }
    const float* pos  = (const float*)d_in[30];
    const float* fc1W = (const float*)d_in[31];
    const float* fc1b = (const float*)d_in[32];
    const float* fc2W = (const float*)d_in[33];
    const float* fc2b = (const float*)d_in[34];
    const int* adj_src = (const int*)d_in[35];
    const int* adj_tgt = (const int*)d_in[36];
    const int* tp_src  = (const int*)d_in[37];
    const int* tp_tgt  = (const int*)d_in[38];
    const float* mask  = (const float*)d_in[39];
    const int* seq     = (const int*)d_in[40];
    const int* uids    = (const int*)d_in[41];
    const int* iids    = (const int*)d_in[42];
    const int* ulocs   = (const int*)d_in[43];
    const int* suids   = (const int*)d_in[44];
    const int* siids   = (const int*)d_in[45];
    long E = (long)in_sizes[35] / G;

    // ---- workspace layout (fp32 offsets) ----
    float* ws = (float*)d_ws;
    size_t off = 0;
    auto alloc = [&](size_t n) { float* p = ws + off; off += n; return p; };
    float* uv    = alloc((size_t)G * U * D);   // GNN user_vector [G,U,D]
    float* iv    = alloc((size_t)G * I * D);   // GNN item_vector [G,I,D]
    float* ut    = alloc((size_t)U * G * D);   // LSTM user outputs [U,4,64]
    float* itb   = alloc((size_t)I * G * D);   // LSTM item outputs [I,4,64]
    float* ucur  = alloc((size_t)U * D);
    float* icur  = alloc((size_t)I * D);
    float* accU  = alloc((size_t)U * D);
    float* accI  = alloc((size_t)I * D);
    float* z     = alloc((size_t)U * ZN);      // LSTM gate pre-activations
    float* hb    = alloc((size_t)U * D);
    float* cb    = alloc((size_t)U * D);
    float* fu    = alloc((size_t)U * D);       // final_u
    float* fi    = alloc((size_t)I * D);       // final_i
    float* att   = alloc((size_t)B * D);
    float* attln = alloc((size_t)B * D);
    float* attmp = alloc((size_t)B * D);
    float* wbuf  = alloc((size_t)U);

    auto nblk = [](long n, int t) { return dim3((unsigned)((n + t - 1) / t)); };
    const long nU = (long)U * D, nI = (long)I * D;

    // ---------------- GNN message passing (4 graphs, 2 layers) -------------
    for (int k = 0; k < G; ++k) {
        k_copy2<<<nblk(nU, 256), 256, 0, stream>>>(ucur, uv + (size_t)k * U * D,
                                                   user_emb + (size_t)k * U * D, nU);
        k_copy2<<<nblk(nI, 256), 256, 0, stream>>>(icur, iv + (size_t)k * I * D,
                                                   item_emb + (size_t)k * I * D, nI);
        for (int layer = 0; layer < 2; ++layer) {
            k_zero<<<nblk(nU, 256), 256, 0, stream>>>(accU, nU);
            k_zero<<<nblk(nI, 256), 256, 0, stream>>>(accI, nI);
            k_scatter<<<nblk(E * 16, 256), 256, 0, stream>>>(
                accU, icur, adj_src + (size_t)k * E, adj_tgt + (size_t)k * E, E);
            k_scatter<<<nblk(E * 16, 256), 256, 0, stream>>>(
                accI, ucur, tp_src + (size_t)k * E, tp_tgt + (size_t)k * E, E);
            k_gnn_update<<<nblk(nU, 256), 256, 0, stream>>>(ucur, uv + (size_t)k * U * D, accU, nU);
            k_gnn_update<<<nblk(nI, 256), 256, 0, stream>>>(icur, iv + (size_t)k * I * D, accI, nI);
        }
    }

    // ---------------- temporal LSTM (fused dual-panel WMMA GEMM) -----------
    // z = bias + x_t @ Wx + h @ Wh   in a single pass over z.
    k_zero<<<nblk(nU, 256), 256, 0, stream>>>(hb, nU);
    k_zero<<<nblk(nU, 256), 256, 0, stream>>>(cb, nU);
    for (int t = 0; t < G; ++t) {
        k_gemm_wmma<<<dim3(ZN / 64, U / 16), 32, 0, stream>>>(
            z, uv + (size_t)t * U * D, l0Wx, hb, l0Wh, l0b, D, D, ZN, D, ZN, ZN);
        k_lstm_gates<<<nblk(nU, 256), 256, 0, stream>>>(z, hb, cb, ut, U, t);
    }
    k_zero<<<nblk(nI, 256), 256, 0, stream>>>(hb, nI);
    k_zero<<<nblk(nI, 256), 256, 0, stream>>>(cb, nI);
    for (int t = 0; t < G; ++t) {
        k_gemm_wmma<<<dim3(ZN / 64, I / 16), 32, 0, stream>>>(
            z, iv + (size_t)t * I * D, l1Wx, hb, l1Wh, l1b, D, D, ZN, D, ZN, ZN);
        k_lstm_gates<<<nblk(nI, 256), 256, 0, stream>>>(z, hb, cb, itb, I, t);
    }

    // ---------------- LN + MHA over the 4 snapshots + mean -----------------
    k_ln_mha_mean<<<U / 4, 128, 0, stream>>>(ut, ln_g[0], ln_b[0], a0Wq, a0Wk, a0Wv, fu, U);
    k_ln_mha_mean<<<I / 4, 128, 0, stream>>>(itb, ln_g[1], ln_b[1], a1Wq, a1Wk, a1Wv, fi, I);

    // ---------------- masked sequence pooling + attn stack -----------------
    k_seq_pool<<<B / 4, 128, 0, stream>>>(mask, seq, fi, pos,
                                          ln_g[2], ln_b[2], ln_g[3], ln_b[3], att, B);
    for (int i = 0; i < 2; ++i) {   // S==1 => mha(x) == x @ Wv
        k_ln_rows<<<B / 4, 128, 0, stream>>>(att, ln_g[4], ln_b[4], attln, B);
        k_gemm_wmma<<<dim3(D / 64, B / 16), 32, 0, stream>>>(
            attmp, attln, asWv[i], nullptr, nullptr, nullptr, D, D, D, 0, 0, D);
        k_leaky_add<<<nblk((long)B * D, 256), 256, 0, stream>>>(att, attmp, (long)B * D);
    }

    // ---------------- predictions + SSL loss -------------------------------
    float* out = (float*)d_out;
    k_preds<<<P / 4, 128, 0, stream>>>(fu, fi, att, uids, iids, ulocs, out, P);
    k_zero<<<1, 32, 0, stream>>>(out + P, 1);
    for (int i = 0; i < G; ++i) {
        k_ssl_w<<<U / 4, 128, 0, stream>>>(fu, uv + (size_t)i * U * D,
                                           fc1W, fc1b, fc2W, fc2b, wbuf, U);
        k_ssl_pair<<<HALF / 4, 128, 0, stream>>>(fu, fi, uv + (size_t)i * U * D,
                                                 iv + (size_t)i * I * D, wbuf,
                                                 suids + (size_t)i * SAMP,
                                                 siids + (size_t)i * SAMP, out + P, HALF);
    }
}